// DiceLoss_15805479649590
// MI455X (gfx1250) — compile-verified
//
#include <hip/hip_runtime.h>

// ---------------------------------------------------------------------------
// DiceLoss on MI455X (gfx1250, wave32)
//   pred:   [8, 19, 512, 512] f32 logits
//   target: [8, 512, 512]     i32 labels
//   out:    scalar f32 = mean over (b,c) of (1 - (2*inter+eps)/(union+eps))
//
// ~167 MB mandatory traffic -> memory-bound (~7us @ 23.3 TB/s).
// Hot loop is tuned to stay under the per-SIMD memory-time budget:
//   - softmax kept in registers, v_rcp instead of IEEE div
//   - union:  1 FMA per channel
//   - inter/counts: per-pixel LDS atomics on bin `t` (no 19-wide select chain)
// Finalize kernel reduces the 152 dice terms with v_wmma_f32_16x16x4_f32.
// ---------------------------------------------------------------------------

typedef __attribute__((ext_vector_type(2))) float v2f;
typedef __attribute__((ext_vector_type(8))) float v8f;

namespace {
constexpr int   kB       = 8;
constexpr int   kC       = 19;
constexpr int   kHW      = 512 * 512;      // 262144 pixels per batch image
constexpr int   kBPB     = 128;            // blocks per batch image
constexpr int   kChunk   = kHW / kBPB;     // 2048 pixels per block
constexpr int   kThreads = 256;            // 8 waves (wave32)
constexpr float kSmooth  = 1e-8f;
}

__global__ __launch_bounds__(kThreads)
void dice_partial_kernel(const float* __restrict__ pred,
                         const int*   __restrict__ target,
                         float*       __restrict__ wsUnion,   // [B*C] (sumP + count)
                         float*       __restrict__ wsInter)   // [B*C]
{
    const int b   = blockIdx.x / kBPB;
    const int blk = blockIdx.x % kBPB;

    const float* predB = pred   + (size_t)b * kC * kHW;
    const int*   targB = target + (size_t)b * kHW;

    __shared__ float        sU[kC];    // block union partials (written at end)
    __shared__ float        sInt[kC];  // block intersection bins
    __shared__ unsigned int sCnt[kC];  // block one-hot counts (target histogram)
    if (threadIdx.x < kC) {
        sU[threadIdx.x]   = 0.0f;
        sInt[threadIdx.x] = 0.0f;
        sCnt[threadIdx.x] = 0u;
    }
    __syncthreads();

    float uAcc[kC];
#pragma unroll
    for (int c = 0; c < kC; ++c) uAcc[c] = 0.0f;

    const int base = blk * kChunk;
    for (int pix = base + (int)threadIdx.x; pix < base + kChunk; pix += kThreads) {
        // 19 fully coalesced (128B/wave) channel loads for this pixel column.
        float x[kC];
#pragma unroll
        for (int c = 0; c < kC; ++c)
            x[c] = predB[(size_t)c * kHW + pix];

        // Numerically-stable softmax in registers.
        float m = x[0];
#pragma unroll
        for (int c = 1; c < kC; ++c) m = fmaxf(m, x[c]);

        float s = 0.0f;
#pragma unroll
        for (int c = 0; c < kC; ++c) { x[c] = __expf(x[c] - m); s += x[c]; }
        const float r = __builtin_amdgcn_rcpf(s);   // 1 trans op, ~1 ulp

        // union partial: one FMA per channel.
#pragma unroll
        for (int c = 0; c < kC; ++c)
            uAcc[c] = fmaf(x[c], r, uAcc[c]);

        // intersection + count for the target channel only.
        // Gather reload of pred[t] hits L0 (same lines as above), and the
        // recomputed exp matches x[t] bit-exactly (same inputs, same ops).
        const int   t  = targB[pix];
        const float xt = predB[(size_t)t * kHW + pix];
        const float pt = __expf(xt - m) * r;
        atomicAdd(&sInt[t], pt);
        atomicAdd(&sCnt[t], 1u);
    }

    // ---- block reduction of union: wave butterfly -> LDS -> global ----
    const int lane = threadIdx.x & 31;
#pragma unroll
    for (int c = 0; c < kC; ++c) {
        float u = uAcc[c];
#pragma unroll
        for (int off = 16; off > 0; off >>= 1)
            u += __shfl_xor(u, off, 32);
        if (lane == 0) atomicAdd(&sU[c], u);
    }
    __syncthreads();

    if (threadIdx.x < kC) {
        const int c = threadIdx.x;
        // union = sum(p) + onehot count, folded here.
        atomicAdd(&wsUnion[b * kC + c], sU[c] + (float)sCnt[c]);
        atomicAdd(&wsInter[b * kC + c], sInt[c]);
    }
}

// Finalize: 152 dice terms, reduced across the wave with a full-precision
// f32 WMMA (A = per-lane partials, B = ones -> D[m][n] = part_m + part_{m+16}).
__global__ __launch_bounds__(32)
void dice_final_kernel(const float* __restrict__ wsUnion,
                       const float* __restrict__ wsInter,
                       float*       __restrict__ out)
{
    const int l = threadIdx.x;   // 0..31, all lanes active (EXEC all ones)

    float partial = 0.0f;
#pragma unroll
    for (int j = 0; j < 5; ++j) {
        const int  idx = l + j * 32;
        const bool ok  = (idx < kB * kC);
        const float U  = ok ? wsUnion[ok ? idx : 0] : 0.0f;
        const float I  = ok ? wsInter[ok ? idx : 0] : 0.0f;
        const float dice = (2.0f * I + kSmooth) / (U + kSmooth);
        partial += ok ? (1.0f - dice) : 0.0f;
    }

    // A-matrix 16x4 f32 layout: lanes 0-15 supply A[m][0..1], lanes 16-31
    // supply A[m][2..3]. partial in K=0/K=2, zeros in K=1/K=3; B = ones
    // => D[m][n] = partial_m + partial_{m+16}.
    v2f a;  a[0] = partial; a[1] = 0.0f;
    v2f bv; bv[0] = 1.0f;   bv[1] = 1.0f;
    v8f cz = {};
    v8f d = __builtin_amdgcn_wmma_f32_16x16x4_f32(
        /*neg_a=*/false, a, /*neg_b=*/false, bv,
        /*c_mod=*/(short)0, cz, /*reuse_a=*/false, /*reuse_b=*/false);

    // Lane 0 holds D[0..7][0], lane 16 holds D[8..15][0]; one xor-16 finishes.
    float s8 = d[0] + d[1] + d[2] + d[3] + d[4] + d[5] + d[6] + d[7];
    const float tot = s8 + __shfl_xor(s8, 16, 32);

    if (l == 0) out[0] = tot * (1.0f / (float)(kB * kC));
}

extern "C" void kernel_launch(void* const* d_in, const int* in_sizes, int n_in,
                              void* d_out, int out_size, void* d_ws, size_t ws_size,
                              hipStream_t stream) {
    const float* pred   = (const float*)d_in[0];
    const int*   target = (const int*)d_in[1];

    float* wsUnion = (float*)d_ws;
    float* wsInter = wsUnion + kB * kC;

    // Zero the 2*152-float accumulator workspace every call (graph-capture safe).
    hipMemsetAsync(d_ws, 0, 2 * kB * kC * sizeof(float), stream);

    dice_partial_kernel<<<kB * kBPB, kThreads, 0, stream>>>(pred, target, wsUnion, wsInter);
    dice_final_kernel<<<1, 32, 0, stream>>>(wsUnion, wsInter, (float*)d_out);
}